// SupervisedInstanceEmbeddingLoss_19413252178519
// MI455X (gfx1250) — compile-verified
//
#include <hip/hip_runtime.h>
#include <hip/hip_bf16.h>
#include <math.h>

// Problem constants (from reference)
#define BB 16
#define NN 65536
#define CC 32
#define HH 1024
#define WW 1024
#define KK 64
#define PUSH_MARGIN 1.0f
#define EPSV 1e-12f

#define SUM_BLOCKS_PER_BATCH 32
#define SUM_THREADS 256

typedef float v2f __attribute__((ext_vector_type(2)));
typedef float v8f __attribute__((ext_vector_type(8)));

// ---------------------------------------------------------------------------
// Kernel 1: gather instance ids:  ids[b,n] = y[b, coords[b,n,1], coords[b,n,0]]
// ---------------------------------------------------------------------------
__global__ void gather_ids_kernel(const int* __restrict__ coords,
                                  const int* __restrict__ y,
                                  int* __restrict__ ids) {
    int tid = blockIdx.x * blockDim.x + threadIdx.x;
    if (tid >= BB * NN) return;
    int b  = tid >> 16;                 // NN == 65536
    int cy = coords[2 * tid];           // coords[...,0] -> column
    int cx = coords[2 * tid + 1];       // coords[...,1] -> row
    int id = y[(size_t)b * HH * WW + (size_t)cx * WW + cy];
    ids[tid] = id & (KK - 1);
}

// ---------------------------------------------------------------------------
// Kernel 2: segment sums + counts via LDS-privatized ds_add_f32.
// Lane-per-channel layout: one coalesced 128B row per sample, conflict-free
// DS atomics (addresses id*32+lane are consecutive across the wave).
// ---------------------------------------------------------------------------
__global__ __launch_bounds__(SUM_THREADS)
void seg_sums_kernel(const float* __restrict__ emb,
                     const int* __restrict__ ids,
                     float* __restrict__ sums,
                     float* __restrict__ counts) {
    __shared__ float lsum[KK * CC];
    __shared__ float lcnt[KK];
    int b   = blockIdx.x / SUM_BLOCKS_PER_BATCH;
    int blk = blockIdx.x % SUM_BLOCKS_PER_BATCH;
    int t   = threadIdx.x;
    for (int i = t; i < KK * CC; i += SUM_THREADS) lsum[i] = 0.0f;
    for (int i = t; i < KK; i += SUM_THREADS) lcnt[i] = 0.0f;
    __syncthreads();

    int lane = t & 31, wave = t >> 5;
    const int spb = NN / SUM_BLOCKS_PER_BATCH;        // samples per block
    const int spw = spb / (SUM_THREADS / 32);         // samples per wave
    int n0 = blk * spb + wave * spw;
    const float* embB = emb + (size_t)b * NN * CC;
    const int*   idsB = ids + (size_t)b * NN;

    for (int s = 0; s < spw; ++s) {
        int n  = n0 + s;
        int id = idsB[n];
        float v = embB[(size_t)n * CC + lane];
        atomicAdd(&lsum[id * CC + lane], v);
        if (lane == 0) atomicAdd(&lcnt[id], 1.0f);
    }
    __syncthreads();

    float* sumsB = sums + (size_t)b * KK * CC;
    for (int i = t; i < KK * CC; i += SUM_THREADS) atomicAdd(&sumsB[i], lsum[i]);
    float* cntB = counts + (size_t)b * KK;
    for (int i = t; i < KK; i += SUM_THREADS) atomicAdd(&cntB[i], lcnt[i]);
}

// ---------------------------------------------------------------------------
// Kernel 3: pull distances. Each block rebuilds centroids in LDS (cheap),
// then per sample: lanes hold channels, wave-shuffle reduce |e - c|^2,
// lane 0 accumulates sqrt into per-id LDS bin.
// ---------------------------------------------------------------------------
__global__ __launch_bounds__(SUM_THREADS)
void pull_dist_kernel(const float* __restrict__ emb,
                      const int* __restrict__ ids,
                      const float* __restrict__ sums,
                      const float* __restrict__ counts,
                      float* __restrict__ pullsum) {
    __shared__ float cent[KK * CC];
    __shared__ float lpull[KK];
    int b   = blockIdx.x / SUM_BLOCKS_PER_BATCH;
    int blk = blockIdx.x % SUM_BLOCKS_PER_BATCH;
    int t   = threadIdx.x;
    for (int i = t; i < KK * CC; i += SUM_THREADS) {
        float cnt = counts[b * KK + i / CC];
        cent[i] = sums[(size_t)b * KK * CC + i] / fmaxf(cnt, 1.0f);
    }
    for (int i = t; i < KK; i += SUM_THREADS) lpull[i] = 0.0f;
    __syncthreads();

    int lane = t & 31, wave = t >> 5;
    const int spb = NN / SUM_BLOCKS_PER_BATCH;
    const int spw = spb / (SUM_THREADS / 32);
    int n0 = blk * spb + wave * spw;
    const float* embB = emb + (size_t)b * NN * CC;
    const int*   idsB = ids + (size_t)b * NN;

    for (int s = 0; s < spw; ++s) {
        int n  = n0 + s;
        int id = idsB[n];
        float diff = embB[(size_t)n * CC + lane] - cent[id * CC + lane];
        float sq = diff * diff;
        // wave32 butterfly reduction
        for (int off = 16; off > 0; off >>= 1) sq += __shfl_xor(sq, off, 32);
        if (lane == 0) atomicAdd(&lpull[id], sqrtf(sq + EPSV));
    }
    __syncthreads();
    for (int i = t; i < KK; i += SUM_THREADS)
        atomicAdd(&pullsum[b * KK + i], lpull[i]);
}

// ---------------------------------------------------------------------------
// Kernel 4: per-batch finalize. Centroid Gram matrix G = Cent * Cent^T via
// V_WMMA_F32_16X16X4_F32 (full f32 precision), then push/pull reduction.
// 128 threads = 4 waves; wave ti handles G row-tile ti, loops 4 col-tiles.
// Uniform control flow -> EXEC all ones at every WMMA.
// ---------------------------------------------------------------------------
__global__ __launch_bounds__(128)
void finalize_kernel(const float* __restrict__ sums,
                     const float* __restrict__ counts,
                     const float* __restrict__ pullsum,
                     float* __restrict__ out) {
    __shared__ float cent[KK * CC];   // 8 KB
    __shared__ float G[KK * KK];      // 16 KB
    __shared__ float red[4];          // push, npairs, pull, ninst
    int b = blockIdx.x;
    int t = threadIdx.x;
    int lane = t & 31, wave = t >> 5;

    for (int i = t; i < KK * CC; i += 128) {
        float cnt = counts[b * KK + i / CC];
        cent[i] = sums[(size_t)b * KK * CC + i] / fmaxf(cnt, 1.0f);
    }
    if (t < 4) red[t] = 0.0f;
    __syncthreads();

    // --- WMMA Gram ---
    int m     = lane & 15;            // M (and N) index within tile
    int kk2   = (lane >> 4) << 1;     // K offset pair: lanes 0-15 -> K 0,1 ; 16-31 -> K 2,3
    int rbase = (lane >> 4) << 3;     // D rows: lanes 16-31 hold M = v + 8
    int ti    = wave;
    for (int tj = 0; tj < 4; ++tj) {
        v8f acc = {};
        for (int s = 0; s < 8; ++s) {            // 8 k-steps of 4 over C=32
            int c0 = 4 * s + kk2;
            v2f a, bmat;
            a.x    = cent[(ti * 16 + m) * CC + c0];
            a.y    = cent[(ti * 16 + m) * CC + c0 + 1];
            bmat.x = cent[(tj * 16 + m) * CC + c0];   // B = Cent^T tile
            bmat.y = cent[(tj * 16 + m) * CC + c0 + 1];
            acc = __builtin_amdgcn_wmma_f32_16x16x4_f32(
                false, a, false, bmat, (short)0, acc, false, false);
        }
#pragma unroll
        for (int v = 0; v < 8; ++v)
            G[(ti * 16 + rbase + v) * KK + (tj * 16 + m)] = acc[v];
    }
    __syncthreads();

    // --- push term over valid upper-triangular pairs ---
    float lpush = 0.0f, lnp = 0.0f;
    for (int idx = t; idx < KK * KK; idx += 128) {
        int i = idx >> 6, j = idx & 63;
        if (i != 0 && i < j) {
            float ci = counts[b * KK + i], cj = counts[b * KK + j];
            if (ci > 0.0f && cj > 0.0f) {
                float d2 = G[i * KK + i] + G[j * KK + j] - 2.0f * G[i * KK + j];
                float dist = sqrtf(fmaxf(d2, 0.0f));
                lpush += fmaxf(PUSH_MARGIN - dist, 0.0f);
                lnp   += 1.0f;
            }
        }
    }
    atomicAdd(&red[0], lpush);
    atomicAdd(&red[1], lnp);

    // --- pull term ---
    if (t < KK) {
        float cnt = counts[b * KK + t];
        if (t != 0 && cnt > 0.0f) {
            atomicAdd(&red[2], pullsum[b * KK + t] / fmaxf(cnt, 1.0f));
            atomicAdd(&red[3], 1.0f);
        }
    }
    __syncthreads();

    if (t == 0) {
        float ninst = red[3];
        if (ninst > 1.0f) {
            float contrib = red[2] / ninst + red[0] / fmaxf(red[1], 1.0f);
            atomicAdd(out, contrib);
        }
    }
}

// ---------------------------------------------------------------------------
extern "C" void kernel_launch(void* const* d_in, const int* in_sizes, int n_in,
                              void* d_out, int out_size, void* d_ws, size_t ws_size,
                              hipStream_t stream) {
    const float* emb    = (const float*)d_in[0];
    const int*   coords = (const int*)d_in[1];
    const int*   y      = (const int*)d_in[2];
    float*       out    = (float*)d_out;

    char*  ws      = (char*)d_ws;
    int*   ids     = (int*)ws;                              // BB*NN ints (4 MB)
    float* sums    = (float*)(ws + (size_t)BB * NN * 4);    // BB*KK*CC floats
    float* counts  = sums + (size_t)BB * KK * CC;           // BB*KK
    float* pullsum = counts + (size_t)BB * KK;              // BB*KK
    (void)in_sizes; (void)n_in; (void)ws_size;

    // zero accumulators + output (graph-capture-safe memset nodes)
    hipMemsetAsync(sums, 0, (size_t)(BB * KK * CC + 2 * BB * KK) * sizeof(float), stream);
    hipMemsetAsync(out, 0, (size_t)out_size * sizeof(float), stream);

    gather_ids_kernel<<<(BB * NN + 255) / 256, 256, 0, stream>>>(coords, y, ids);
    seg_sums_kernel<<<BB * SUM_BLOCKS_PER_BATCH, SUM_THREADS, 0, stream>>>(emb, ids, sums, counts);
    pull_dist_kernel<<<BB * SUM_BLOCKS_PER_BATCH, SUM_THREADS, 0, stream>>>(emb, ids, sums, counts, pullsum);
    finalize_kernel<<<BB, 128, 0, stream>>>(sums, counts, pullsum, out);
}